// GCN_STUDENT_ENSAMBLE_73873437491479
// MI455X (gfx1250) — compile-verified
//
#include <hip/hip_runtime.h>
#include <hip/hip_bf16.h>

typedef __attribute__((ext_vector_type(2))) float v2f;
typedef __attribute__((ext_vector_type(8))) float v8f;
typedef __attribute__((ext_vector_type(4))) int   v4i;

#define GCN_N   8192
#define GCN_NC  8
#define KSPLIT  8
#define KCHUNK  (GCN_N / KSPLIT)   // 1024 rows of B per block
#define WPB     8                  // waves per block (256 threads, wave32)
#define TILES   (GCN_N / 16)       // 512 row tiles of 16

// gfx1250 async global->LDS copy path (guarded; fallback = vector loads)
#if defined(__has_builtin)
#if __has_builtin(__builtin_amdgcn_global_load_async_to_lds_b128) && \
    __has_builtin(__builtin_amdgcn_s_wait_asynccnt)
#define HAVE_ASYNC_LDS 1
#endif
#endif
#ifndef HAVE_ASYNC_LDS
#define HAVE_ASYNC_LDS 0
#endif

#if HAVE_ASYNC_LDS
typedef v4i __attribute__((address_space(1)))* as1_v4i_ptr;   // global int4*
typedef v4i __attribute__((address_space(3)))* as3_v4i_ptr;   // LDS    int4*
#endif

// ---------------------------------------------------------------------------
// Skinny GEMM: Out[m, c] += sum_k A[m,k] * B[k,c]   (c in [0,8), padded to 16)
// Block = 8 row-tiles sharing one K-chunk; B chunk staged in LDS (async copy).
// f32 WMMA 16x16x4 accumulation; split-K partials via global f32 atomics.
// A-layout (ISA 7.12.2, 32-bit A 16x4): lane L<16 -> (M=L, K=0..1) in v0/v1,
// lanes 16..31 -> (M=L-16, K=2..3). B symmetric: lane holds B[K=2*hi+{0,1}, N=m].
// C layout: VGPR r: lanes 0-15 = (M=r, N=lane), lanes 16-31 = (M=8+r, N=lane-16).
// ---------------------------------------------------------------------------
__global__ __launch_bounds__(256) void gcn_skinny_gemm(const float* __restrict__ A,
                                                       const float* __restrict__ B,
                                                       float* __restrict__ Out) {
  __shared__ float sB[KCHUNK * GCN_NC];          // 32 KB
  const int tid  = threadIdx.x;
  const int lane = tid & 31;
  const int wave = tid >> 5;
  const int kc   = blockIdx.x % KSPLIT;
  const int tile = (blockIdx.x / KSPLIT) * WPB + wave;
  const int kbeg = kc * KCHUNK;

  // ---- stage B[kbeg : kbeg+KCHUNK, 0:8] into LDS, 16B per lane per op ----
  const float* __restrict__ gsrc = B + (size_t)kbeg * GCN_NC;
#if HAVE_ASYNC_LDS
#pragma unroll
  for (int t = 0; t < (KCHUNK * GCN_NC) / (256 * 4); ++t) {   // 8 iters
    const int i = t * 256 + tid;                              // 16B chunk idx
    __builtin_amdgcn_global_load_async_to_lds_b128(
        (as1_v4i_ptr)(gsrc + i * 4),
        (as3_v4i_ptr)(sB + i * 4),
        /*offset=*/0, /*cpol=*/0);
  }
  __builtin_amdgcn_s_wait_asynccnt(0);
#else
#pragma unroll
  for (int t = 0; t < (KCHUNK * GCN_NC) / (256 * 4); ++t) {
    const int i = t * 256 + tid;
    *(float4*)(sB + i * 4) = *(const float4*)(gsrc + i * 4);
  }
#endif
  __syncthreads();

  const int row0 = tile * 16;
  const int m    = lane & 15;
  const int hi   = lane >> 4;
  const float bmask = (m < GCN_NC) ? 1.0f : 0.0f;   // zero-pad B cols 8..15
  const int   nn    = m & (GCN_NC - 1);             // clamped col, always in-bounds

  const float* __restrict__ Arow = A + (size_t)(row0 + m) * GCN_N + kbeg;

  v8f acc = {};
  for (int k0 = 0; k0 < KCHUNK; k0 += 16) {
    __builtin_prefetch(Arow + k0 + 2 * hi + 128, 0, 3);       // global_prefetch_b8
#pragma unroll
    for (int u = 0; u < 4; ++u) {
      const int kk = k0 + 4 * u + 2 * hi;
      v2f a = *(const v2f*)(Arow + kk);                       // A[m, kbeg+kk .. +1]
      v2f b;
      b.x = sB[kk * GCN_NC + nn] * bmask;                     // B[kbeg+kk,   n]
      b.y = sB[(kk + 1) * GCN_NC + nn] * bmask;               // B[kbeg+kk+1, n]
      acc = __builtin_amdgcn_wmma_f32_16x16x4_f32(
          /*neg_a=*/false, a, /*neg_b=*/false, b,
          /*c_mod=*/(short)0, acc, /*reuse_a=*/false, /*reuse_b=*/false);
    }
  }

  // split-K partial accumulate (Out zero-initialized by host-side memset)
  if (m < GCN_NC) {
#pragma unroll
    for (int r = 0; r < 8; ++r) {
      atomicAdd(&Out[(size_t)(row0 + r + 8 * hi) * GCN_NC + m], acc[r]);
    }
  }
}

// ---------------------------------------------------------------------------
// node_embeddings = relu(raw + gc_b), elementwise over N*8
// ---------------------------------------------------------------------------
__global__ __launch_bounds__(256) void gcn_bias_relu(const float* __restrict__ raw,
                                                     const float* __restrict__ gc_b,
                                                     float* __restrict__ emb) {
  const int i = blockIdx.x * 256 + threadIdx.x;
  float v = raw[i] + gc_b[i & (GCN_NC - 1)];
  emb[i] = v > 0.0f ? v : 0.0f;
}

// ---------------------------------------------------------------------------
// Per-row log_softmax, weighted by lin_w[row], block-level LDS tree reduction.
// partial[blk*8 + c] = sum_{rows in blk} (emb[row,c] - lse(row)) * lin_w[row]
// ---------------------------------------------------------------------------
__global__ __launch_bounds__(256) void gcn_lsm_partial(const float* __restrict__ emb,
                                                       const float* __restrict__ lin_w,
                                                       float* __restrict__ partial) {
  __shared__ float sdata[256 * GCN_NC];
  const int tid = threadIdx.x;
  const int row = blockIdx.x * 256 + tid;

  float e[GCN_NC];
  float mx = -3.402823466e38f;
#pragma unroll
  for (int c = 0; c < GCN_NC; ++c) {
    e[c] = emb[(size_t)row * GCN_NC + c];
    mx = e[c] > mx ? e[c] : mx;
  }
  float s = 0.0f;
#pragma unroll
  for (int c = 0; c < GCN_NC; ++c) s += __expf(e[c] - mx);
  const float lse = mx + __logf(s);
  const float w = lin_w[row];
#pragma unroll
  for (int c = 0; c < GCN_NC; ++c) sdata[tid * GCN_NC + c] = (e[c] - lse) * w;
  __syncthreads();

  for (int stride = 128; stride > 0; stride >>= 1) {
    if (tid < stride) {
#pragma unroll
      for (int c = 0; c < GCN_NC; ++c)
        sdata[tid * GCN_NC + c] += sdata[(tid + stride) * GCN_NC + c];
    }
    __syncthreads();
  }
  if (tid < GCN_NC) partial[blockIdx.x * GCN_NC + tid] = sdata[tid];
}

// ---------------------------------------------------------------------------
// out[c] = lin_b + sum over block partials
// ---------------------------------------------------------------------------
__global__ void gcn_final(const float* __restrict__ partial,
                          const float* __restrict__ lin_b,
                          float* __restrict__ out, int nblk) {
  const int c = threadIdx.x;
  if (c < GCN_NC) {
    float s = lin_b[0];
    for (int b = 0; b < nblk; ++b) s += partial[b * GCN_NC + c];
    out[c] = s;
  }
}

// ---------------------------------------------------------------------------
extern "C" void kernel_launch(void* const* d_in, const int* in_sizes, int n_in,
                              void* d_out, int out_size, void* d_ws, size_t ws_size,
                              hipStream_t stream) {
  const float* x     = (const float*)d_in[0];
  const float* adj   = (const float*)d_in[1];
  const float* gc_w  = (const float*)d_in[2];
  const float* gc_b  = (const float*)d_in[3];
  const float* lin_w = (const float*)d_in[4];
  const float* lin_b = (const float*)d_in[5];

  float* out     = (float*)d_out;                 // [0:8] scores
  float* emb     = out + GCN_NC;                  // [8 : 8+65536] node_embeddings
  float* support = (float*)d_ws;                  // 65536 f32
  float* raw2    = support + GCN_N * GCN_NC;      // 65536 f32
  float* partial = raw2 + GCN_N * GCN_NC;         // 32*8 f32

  // zero the atomic-accumulate buffers every call (capturable memset node)
  (void)hipMemsetAsync(d_ws, 0, (size_t)2 * GCN_N * GCN_NC * sizeof(float), stream);

  const dim3 blk(256);
  const dim3 gemm_grid((TILES / WPB) * KSPLIT);   // 512 blocks = 4096 waves

  gcn_skinny_gemm<<<gemm_grid, blk, 0, stream>>>(x,   gc_w,    support);
  gcn_skinny_gemm<<<gemm_grid, blk, 0, stream>>>(adj, support, raw2);
  gcn_bias_relu  <<<dim3(GCN_N * GCN_NC / 256), blk, 0, stream>>>(raw2, gc_b, emb);
  gcn_lsm_partial<<<dim3(GCN_N / 256), blk, 0, stream>>>(emb, lin_w, partial);
  gcn_final      <<<dim3(1), dim3(64), 0, stream>>>(partial, lin_b, out, GCN_N / 256);
}